// DMOptimizer_81449759801976
// MI455X (gfx1250) — compile-verified
//
#include <hip/hip_runtime.h>

// 2-layer LSTM (single step) over N independent scalar streams, H=20.
// Memory-bound (~1.3 GB traffic); math done with V_WMMA_F32_16X16X4_F32
// (fp32 matrix core, K=20 = 5 steps of 4, gate dim 80 = 5 tiles of 16).

typedef float v2f __attribute__((ext_vector_type(2)));
typedef float v8f __attribute__((ext_vector_type(8)));

#define HS   20      // hidden size
#define GW   80      // 4*HS gate width
#define ZROW 84      // padded z row: 8*84 % 64 != 0 -> no LDS bank conflict between lane halves
#define WPB  8       // waves per block (wave32)

__device__ __forceinline__ float fsig(float x) { return 1.0f / (1.0f + __expf(-x)); }

__global__ __launch_bounds__(WPB * 32)
void lstm2_wmma(const float* __restrict__ xin,
                const float* __restrict__ h0p, const float* __restrict__ c0p,
                const float* __restrict__ h1p, const float* __restrict__ c1p,
                const float* __restrict__ W0,  const float* __restrict__ U0,
                const float* __restrict__ b0,
                const float* __restrict__ W1,  const float* __restrict__ U1,
                const float* __restrict__ b1,
                const float* __restrict__ Wd,  const float* __restrict__ bd,
                float* __restrict__ out,
                float* __restrict__ h0n, float* __restrict__ c0n,
                float* __restrict__ h1n, float* __restrict__ c1n,
                int ngroups)
{
    __shared__ float z_lds[WPB * 16 * ZROW];   // 43008 B
    __shared__ float ha_lds[WPB * 16 * HS];    // 10240 B
    __shared__ float hb_lds[WPB * 16 * HS];    // 10240 B
    __shared__ float x_lds[WPB * 16];          //   512 B

    const int lane = threadIdx.x & 31;
    const int wid  = threadIdx.x >> 5;
    int group = blockIdx.x * WPB + wid;
    if (group >= ngroups) group = ngroups - 1;   // tail waves duplicate last group (same data -> same writes)

    float* zb = z_lds  + wid * 16 * ZROW;
    float* hA = ha_lds + wid * 16 * HS;
    float* hB = hb_lds + wid * 16 * HS;
    float* xb = x_lds  + wid * 16;

    const int half = lane >> 4;            // 0: lanes 0-15, 1: lanes 16-31
    const int r    = lane & 15;
    const size_t eb    = (size_t)group * 16;   // first element of this wave's group
    const size_t hbase = eb * HS;              // base offset into [N,20] arrays

    // ---- stage 0: coalesced preload of x and prev-h0 into LDS ----
    if (lane < 16) xb[lane] = xin[eb + lane];
    #pragma unroll
    for (int t = 0; t < 10; ++t) hA[lane + 32 * t] = h0p[hbase + lane + 32 * t];
    __syncthreads();

    // ---- layer 0: z0 = x*W0 + h0 @ U0 + b0  (WMMA f32 16x16x4) ----
    // A 16x4 layout: lane r = row r; lanes 0-15 hold K={k,k+1}, lanes 16-31 K={k+2,k+3}
    v2f a0[5];
    #pragma unroll
    for (int kt = 0; kt < 5; ++kt) {
        const int k = 4 * kt + 2 * half;
        a0[kt].x = hA[r * HS + k];
        a0[kt].y = hA[r * HS + k + 1];
    }
    #pragma unroll
    for (int j = 0; j < 5; ++j) {
        const int c = 16 * j + r;          // gate column; B lane r = column c, K split like A
        v8f acc = {};
        #pragma unroll
        for (int kt = 0; kt < 5; ++kt) {
            const int k = 4 * kt + 2 * half;
            v2f b;
            b.x = U0[k * GW + c];
            b.y = U0[(k + 1) * GW + c];
            acc = __builtin_amdgcn_wmma_f32_16x16x4_f32(false, a0[kt], false, b,
                                                        (short)0, acc, false, false);
        }
        const float w0c = W0[c], b0c = b0[c];
        #pragma unroll
        for (int v = 0; v < 8; ++v) {      // D: lane r holds (M = v+8*half, N = c)
            const int m = v + 8 * half;
            zb[m * ZROW + c] = acc[v] + xb[m] * w0c + b0c;
        }
    }
    __syncthreads();

    // ---- gates layer 0 (p = elem*HS+idx runs contiguously -> coalesced global I/O) ----
    #pragma unroll
    for (int t = 0; t < 10; ++t) {
        const int p = lane + 32 * t;
        const int m = p / HS, idx = p % HS;
        const float* zr = zb + m * ZROW;
        const float ig = fsig(zr[idx]);
        const float fg = fsig(zr[HS + idx]);
        const float gg = tanhf(zr[2 * HS + idx]);
        const float og = fsig(zr[3 * HS + idx]);
        const float cn = fg * c0p[hbase + p] + ig * gg;
        const float hn = og * tanhf(cn);
        c0n[hbase + p] = cn;
        h0n[hbase + p] = hn;
        hB[p] = hn;                        // keep h0_new for layer-1 A matrix
        hA[p] = h1p[hbase + p];            // coalesced prefetch of prev-h1
    }
    __syncthreads();

    // ---- layer 1: z1 = h0n @ W1 + h1 @ U1 + b1 ----
    v2f aW[5], aU[5];
    #pragma unroll
    for (int kt = 0; kt < 5; ++kt) {
        const int k = 4 * kt + 2 * half;
        aW[kt].x = hB[r * HS + k];  aW[kt].y = hB[r * HS + k + 1];
        aU[kt].x = hA[r * HS + k];  aU[kt].y = hA[r * HS + k + 1];
    }
    #pragma unroll
    for (int j = 0; j < 5; ++j) {
        const int c = 16 * j + r;
        v8f acc = {};
        #pragma unroll
        for (int kt = 0; kt < 5; ++kt) {
            const int k = 4 * kt + 2 * half;
            v2f b;
            b.x = W1[k * GW + c];  b.y = W1[(k + 1) * GW + c];
            acc = __builtin_amdgcn_wmma_f32_16x16x4_f32(false, aW[kt], false, b,
                                                        (short)0, acc, false, false);
            b.x = U1[k * GW + c];  b.y = U1[(k + 1) * GW + c];
            acc = __builtin_amdgcn_wmma_f32_16x16x4_f32(false, aU[kt], false, b,
                                                        (short)0, acc, false, false);
        }
        const float b1c = b1[c];
        #pragma unroll
        for (int v = 0; v < 8; ++v)
            zb[(v + 8 * half) * ZROW + c] = acc[v] + b1c;
    }
    __syncthreads();

    // ---- gates layer 1 ----
    #pragma unroll
    for (int t = 0; t < 10; ++t) {
        const int p = lane + 32 * t;
        const int m = p / HS, idx = p % HS;
        const float* zr = zb + m * ZROW;
        const float ig = fsig(zr[idx]);
        const float fg = fsig(zr[HS + idx]);
        const float gg = tanhf(zr[2 * HS + idx]);
        const float og = fsig(zr[3 * HS + idx]);
        const float cn = fg * c1p[hbase + p] + ig * gg;
        const float hn = og * tanhf(cn);
        c1n[hbase + p] = cn;
        h1n[hbase + p] = hn;
        hB[p] = hn;                        // h1_new for dense head
    }
    __syncthreads();

    // ---- dense head: out = h1n @ Wd + bd ----
    if (lane < 16) {
        float acc = bd[0];
        #pragma unroll
        for (int k = 0; k < HS; ++k) acc += hB[lane * HS + k] * Wd[k];
        out[eb + lane] = acc;
    }
}

extern "C" void kernel_launch(void* const* d_in, const int* in_sizes, int n_in,
                              void* d_out, int out_size, void* d_ws, size_t ws_size,
                              hipStream_t stream) {
    (void)n_in; (void)out_size; (void)d_ws; (void)ws_size;
    const float* xin = (const float*)d_in[0];
    const float* h0p = (const float*)d_in[1];
    const float* c0p = (const float*)d_in[2];
    const float* h1p = (const float*)d_in[3];
    const float* c1p = (const float*)d_in[4];
    const float* W0  = (const float*)d_in[5];
    const float* U0  = (const float*)d_in[6];
    const float* b0  = (const float*)d_in[7];
    const float* W1  = (const float*)d_in[8];
    const float* U1  = (const float*)d_in[9];
    const float* b1  = (const float*)d_in[10];
    const float* Wd  = (const float*)d_in[11];
    const float* bd  = (const float*)d_in[12];

    float* out = (float*)d_out;
    const int N = in_sizes[0];                 // 2048*1024 = 2^21
    float* h0n = out + (size_t)N;              // outputs concatenated in return order
    float* c0n = h0n + (size_t)N * HS;
    float* h1n = c0n + (size_t)N * HS;
    float* c1n = h1n + (size_t)N * HS;

    const int ngroups = (N + 15) / 16;         // 16 elements per wave
    const int blocks  = (ngroups + WPB - 1) / WPB;
    lstm2_wmma<<<blocks, WPB * 32, 0, stream>>>(xin, h0p, c0p, h1p, c1p,
                                                W0, U0, b0, W1, U1, b1, Wd, bd,
                                                out, h0n, c0n, h1n, c1n, ngroups);
}